// LNN_14345190769017
// MI455X (gfx1250) — compile-verified
//
#include <hip/hip_runtime.h>

typedef float v2f __attribute__((ext_vector_type(2)));
typedef float v8f __attribute__((ext_vector_type(8)));
typedef int   v4i __attribute__((vector_size(4 * sizeof(int))));

#define DT     0.01f
#define ALPHA  0.5f
#define BETA   0.01f
#define ETA    0.01f
#define T_STEPS 64
#define NB  16384
#define DIN 64
#define HD  128

#if defined(__HIP_DEVICE_COMPILE__) && __has_builtin(__builtin_amdgcn_global_load_async_to_lds_b128)
#define HAVE_ASYNC_LDS 1
#else
#define HAVE_ASYNC_LDS 0
#endif

static __device__ __forceinline__ v8f wmma_f32_k4(v2f a, v2f b, v8f c) {
  // D = A(16x4) * B(4x16) + C(16x16), all fp32
  return __builtin_amdgcn_wmma_f32_16x16x4_f32(false, a, false, b, (short)0, c,
                                               false, false);
}

#if HAVE_ASYNC_LDS
typedef __attribute__((address_space(1))) v4i* gptr_v4i;
typedef __attribute__((address_space(3))) v4i* lptr_v4i;
static __device__ __forceinline__ void async_copy16(const void* gsrc, void* ldst) {
  __builtin_amdgcn_global_load_async_to_lds_b128((gptr_v4i)gsrc, (lptr_v4i)ldst,
                                                 0, 0);
}
static __device__ __forceinline__ void async_wait0() {
#if __has_builtin(__builtin_amdgcn_s_wait_asynccnt)
  __builtin_amdgcn_s_wait_asynccnt(0);
#else
  asm volatile("s_wait_asynccnt 0x0" ::: "memory");
#endif
}
#endif

// Pair-packed W_t layout: Wp[((k>>1)*HD + n)*2 + (k&1)] == W[k][n]
static __device__ __forceinline__ int wp_idx(int row, int col) {
  return ((row >> 1) * HD + col) * 2 + (row & 1);
}

// ---------------------------------------------------------------------------
// init: h = 0, Wt(packed) = W
// ---------------------------------------------------------------------------
__global__ void lnn_init(const float* __restrict__ W, float* __restrict__ h,
                         float* __restrict__ Wt) {
  int idx = blockIdx.x * blockDim.x + threadIdx.x;
  if (idx < NB * HD) h[idx] = 0.0f;
  if (idx < HD * HD) {
    int row = idx / HD, col = idx % HD;
    Wt[wp_idx(row, col)] = W[idx];
  }
}

// ---------------------------------------------------------------------------
// xU = x @ U + b   ([16384,64] @ [64,128]), WMMA f32 16x16x4
// block = 256 threads (8 waves), each block does 64 rows; wave -> 4 col tiles
// ---------------------------------------------------------------------------
__global__ void lnn_xu(const float* __restrict__ x, const float* __restrict__ U,
                       const float* __restrict__ bvec, float* __restrict__ xU) {
  __shared__ float sU[DIN * HD];   // 32 KB
  __shared__ float sx[64 * DIN];   // 16 KB
  int tid = threadIdx.x;
  int rowBase = blockIdx.x * 64;
  for (int i = tid; i < DIN * HD; i += 256) sU[i] = U[i];
  for (int i = tid; i < 64 * DIN; i += 256) sx[i] = x[rowBase * DIN + i];
  __syncthreads();

  int wave = tid >> 5, lane = tid & 31;
  int rowTile = wave & 3;        // 4 row tiles of 16
  int colGroup = wave >> 2;      // 2 groups x 4 col tiles
  int lm = lane & 15;
  int lk = (lane >> 4) * 2;

  v8f acc[4] = {};
  for (int k = 0; k < DIN; k += 4) {
    v2f a;
    int ar = rowTile * 16 + lm;
    a.x = sx[ar * DIN + k + lk];
    a.y = sx[ar * DIN + k + lk + 1];
    for (int c = 0; c < 4; ++c) {
      int cn = (colGroup * 4 + c) * 16 + lm;
      v2f b;
      b.x = sU[(k + lk) * HD + cn];
      b.y = sU[(k + lk + 1) * HD + cn];
      acc[c] = wmma_f32_k4(a, b, acc[c]);
    }
  }
  for (int c = 0; c < 4; ++c) {
    int col = (colGroup * 4 + c) * 16 + lm;
    for (int r = 0; r < 8; ++r) {
      int m = rowTile * 16 + r + ((lane >> 4) * 8);   // D layout: M = r (+8 hi lanes)
      xU[(rowBase + m) * HD + col] = acc[c][r] + bvec[col];
    }
  }
}

// ---------------------------------------------------------------------------
// h = h + DT*(-ALPHA*h + tanh(h @ Wt + xU))
// Wt (pair-packed, 64 KB) + h tile (32 KB) staged in LDS via async-to-LDS
// (ASYNCcnt path) when available; B fragment = single aligned ds_load_b64.
// ---------------------------------------------------------------------------
__global__ void lnn_step_h(const float* __restrict__ xU, float* __restrict__ h,
                           const float* __restrict__ Wt) {
  __shared__ float sW[HD * HD];   // 64 KB, pair-packed (same layout as Wt)
  __shared__ float sh[64 * HD];   // 32 KB  (old h, block-private rows)
  int tid = threadIdx.x;
  int rowBase = blockIdx.x * 64;

#if HAVE_ASYNC_LDS
  // 16B per thread per iteration; identity copy global->LDS
  for (int i = tid; i < (HD * HD) / 4; i += 256)
    async_copy16(Wt + i * 4, sW + i * 4);
  for (int i = tid; i < (64 * HD) / 4; i += 256)
    async_copy16(h + (size_t)rowBase * HD + i * 4, sh + i * 4);
  async_wait0();
#else
  for (int i = tid; i < HD * HD; i += 256) sW[i] = Wt[i];
  for (int i = tid; i < 64 * HD; i += 256) sh[i] = h[rowBase * HD + i];
#endif
  __syncthreads();

  int wave = tid >> 5, lane = tid & 31;
  int rowTile = wave & 3;
  int colGroup = wave >> 2;
  int lm = lane & 15;
  int lk = (lane >> 4) * 2;

  v8f acc[4] = {};
  for (int k = 0; k < HD; k += 4) {
    v2f a;
    int ar = rowTile * 16 + lm;
    a.x = sh[ar * HD + k + lk];
    a.y = sh[ar * HD + k + lk + 1];
    int prow = ((k + lk) >> 1) * HD * 2;   // packed row-pair base (k+lk is even)
    for (int c = 0; c < 4; ++c) {
      int cn = (colGroup * 4 + c) * 16 + lm;
      v2f b;
      b.x = sW[prow + cn * 2];       // W[k+lk  ][cn]
      b.y = sW[prow + cn * 2 + 1];   // W[k+lk+1][cn]  (adjacent -> ds_load_b64)
      acc[c] = wmma_f32_k4(a, b, acc[c]);
    }
  }
  for (int c = 0; c < 4; ++c) {
    int col = (colGroup * 4 + c) * 16 + lm;
    for (int r = 0; r < 8; ++r) {
      int m = rowTile * 16 + r + ((lane >> 4) * 8);
      float hv = sh[m * HD + col];                 // old h from LDS (hazard-free)
      float pre = acc[c][r] + xU[(rowBase + m) * HD + col];
      h[(rowBase + m) * HD + col] = hv + DT * (-ALPHA * hv + tanhf(pre));
    }
  }
}

// ---------------------------------------------------------------------------
// G = h^T @ h (full batch);  Wt += DT*(ETA*G/NB - BETA*Wt)   (packed Wt)
// 64 blocks = 8x8 tiles of 16x16; 8 waves split K=16384; deterministic
// fixed-order LDS reduction; 2 interleaved accumulators break WMMA RAW chain
// ---------------------------------------------------------------------------
__global__ void lnn_step_w(const float* __restrict__ h, float* __restrict__ Wt) {
  __shared__ float sp[8 * 32 * 8];   // 8 KB of wave partials
  int tid = threadIdx.x;
  int wave = tid >> 5, lane = tid & 31;
  int ti = blockIdx.x >> 3;
  int tj = blockIdx.x & 7;
  int lm = lane & 15;
  int lk = (lane >> 4) * 2;
  int ci = ti * 16 + lm;
  int cj = tj * 16 + lm;

  v8f acc0 = {}, acc1 = {};
  int kBase = wave * (NB / 8);
  for (int k = kBase; k < kBase + NB / 8; k += 8) {
    {
      int kr = k + lk;
      v2f a, b;
      a.x = h[kr * HD + ci];       a.y = h[(kr + 1) * HD + ci];
      b.x = h[kr * HD + cj];       b.y = h[(kr + 1) * HD + cj];
      acc0 = wmma_f32_k4(a, b, acc0);
    }
    {
      int kr = k + 4 + lk;
      v2f a, b;
      a.x = h[kr * HD + ci];       a.y = h[(kr + 1) * HD + ci];
      b.x = h[kr * HD + cj];       b.y = h[(kr + 1) * HD + cj];
      acc1 = wmma_f32_k4(a, b, acc1);
    }
  }
  v8f acc = acc0 + acc1;
  for (int r = 0; r < 8; ++r) sp[(wave * 32 + lane) * 8 + r] = acc[r];
  __syncthreads();

  // one thread per element of the 16x16 tile; fixed-order sum over 8 waves
  int m = tid >> 4;
  int n = tid & 15;
  int lsrc = n + 16 * (m >> 3);    // lane holding D[m][n]
  int rsrc = m & 7;                // vgpr index holding D[m][n]
  float g = 0.0f;
  for (int w = 0; w < 8; ++w) g += sp[(w * 32 + lsrc) * 8 + rsrc];
  int row = ti * 16 + m;
  int col = tj * 16 + n;
  int pi = wp_idx(row, col);
  float wv = Wt[pi];
  Wt[pi] = wv + DT * (ETA * g / (float)NB - BETA * wv);
}

// ---------------------------------------------------------------------------
// out = h @ w_out^T + b_out   (D_OUT = 1)
// ---------------------------------------------------------------------------
__global__ void lnn_out(const float* __restrict__ h, const float* __restrict__ w_out,
                        const float* __restrict__ b_out, float* __restrict__ out) {
  __shared__ float sw[HD];
  int tid = threadIdx.x;
  if (tid < HD) sw[tid] = w_out[tid];
  __syncthreads();
  int row = blockIdx.x * blockDim.x + tid;
  float s = 0.0f;
  for (int k = 0; k < HD; ++k) s += h[row * HD + k] * sw[k];
  out[row] = s + b_out[0];
}

extern "C" void kernel_launch(void* const* d_in, const int* in_sizes, int n_in,
                              void* d_out, int out_size, void* d_ws, size_t ws_size,
                              hipStream_t stream) {
  const float* x     = (const float*)d_in[0];
  const float* W     = (const float*)d_in[1];
  const float* U     = (const float*)d_in[2];
  const float* b     = (const float*)d_in[3];
  const float* w_out = (const float*)d_in[4];
  const float* b_out = (const float*)d_in[5];
  float* out = (float*)d_out;

  float* ws = (float*)d_ws;
  float* xU = ws;                          // 16384*128 f32 (8 MB)
  float* h  = xU + (size_t)NB * HD;        // 16384*128 f32 (8 MB)
  float* Wt = h  + (size_t)NB * HD;        // 128*128  f32 (64 KB, pair-packed)

  lnn_init<<<(NB * HD + 255) / 256, 256, 0, stream>>>(W, h, Wt);
  lnn_xu<<<NB / 64, 256, 0, stream>>>(x, U, b, xU);
  for (int t = 0; t < T_STEPS; ++t) {
    lnn_step_h<<<NB / 64, 256, 0, stream>>>(xU, h, Wt);
    lnn_step_w<<<64, 256, 0, stream>>>(h, Wt);
  }
  lnn_out<<<NB / 256, 256, 0, stream>>>(h, w_out, b_out, out);
}